// CommNet_25288767439260
// MI455X (gfx1250) — compile-verified
//
#include <hip/hip_runtime.h>
#include <hip/hip_bf16.h>

// ---------------------------------------------------------------------------
// CommNet forward on gfx1250 (wave32, WMMA fp32 16x16x4).
// Rows bucketed by model id -> grouped GEMMs on the matrix units.
// Each wave owns a 16x64 C tile: 1 A fragment feeds 4 independent WMMA
// accumulator chains per K-step (XDL overlap, 4x less A traffic).
// Weight set (~152 MB) is L2-resident (192 MB L2); WGP-scope prefetch
// (inline asm, SCOPE 0 = pull into all cache levels) stages upcoming
// weight rows L2 -> WGP$ ahead of the strided B walk.
// ---------------------------------------------------------------------------

typedef __attribute__((ext_vector_type(2))) float v2f;
typedef __attribute__((ext_vector_type(8))) float v8f;

#define HIDSZ    256
#define NAGENTS  8
#define NMODELS  64
#define NACTIONS 16
#define BN       256
#define IN_DIM   512
#define COMMSZ   (HIDSZ * NAGENTS)   // 2048

static __device__ __forceinline__ v8f wmma_f32(v2f a, v2f b, v8f c) {
  // (neg_a, A, neg_b, B, c_mod, C, reuse_a, reuse_b)
  return __builtin_amdgcn_wmma_f32_16x16x4_f32(false, a, false, b, (short)0, c,
                                               false, false);
}

// WGP-scope prefetch: no scope modifier == SCOPE 0 (WGP) -> fills WGP$.
static __device__ __forceinline__ void prefetch_wgp(const void* p) {
  asm volatile("global_prefetch_b8 %0, off" ::"v"(p));
}

// --------------------------------------------------------------------------
// Kernel 1: bucket rows by model id. Single 256-thread block.
// --------------------------------------------------------------------------
__global__ __launch_bounds__(BN) void group_kernel(const int* __restrict__ ids,
                                                   int* __restrict__ counts,
                                                   int* __restrict__ rowidx) {
  int t = threadIdx.x;
  if (t < NMODELS) counts[t] = 0;
  __syncthreads();
  int m = ids[t];
  int pos = atomicAdd(&counts[m], 1);
  rowidx[m * BN + pos] = t;
}

// --------------------------------------------------------------------------
// Kernel 2: pre = inp @ enc_w + enc_b + sum_a comm_in   (dense WMMA GEMM)
// One wave per 16x64 C tile; K loop over 512 in steps of 4.
// --------------------------------------------------------------------------
__global__ __launch_bounds__(32) void enc_pre_kernel(
    const float* __restrict__ inp,      // [BN, IN_DIM]
    const float* __restrict__ enc_w,    // [IN_DIM, HIDSZ]
    const float* __restrict__ enc_b,    // [HIDSZ]
    const float* __restrict__ comm_in,  // [BN, NAGENTS, HIDSZ]
    float* __restrict__ pre)            // [BN, HIDSZ]
{
  const int m0 = (blockIdx.x >> 2) << 4;     // row tile base (16 tiles)
  const int n0 = (blockIdx.x & 3) << 6;      // col group base (4 x 64)
  const int lane = threadIdx.x;
  const int up = lane >> 4;                  // 0: K pair lo, 1: K pair hi
  const int l  = lane & 15;

  v8f c0 = {}, c1 = {}, c2 = {}, c3 = {};
#pragma unroll 4
  for (int k = 0; k < IN_DIM; k += 4) {
    const int kk = k + 2 * up;
    if ((k & 31) == 0 && k + 32 < IN_DIM)
      prefetch_wgp(&enc_w[(kk + 32) * HIDSZ + n0]);
    const v2f a = *(const v2f*)&inp[(m0 + l) * IN_DIM + kk];
    v2f b;
    b.x = enc_w[(kk + 0) * HIDSZ + n0 + l];
    b.y = enc_w[(kk + 1) * HIDSZ + n0 + l];
    c0 = wmma_f32(a, b, c0);
    b.x = enc_w[(kk + 0) * HIDSZ + n0 + 16 + l];
    b.y = enc_w[(kk + 1) * HIDSZ + n0 + 16 + l];
    c1 = wmma_f32(a, b, c1);
    b.x = enc_w[(kk + 0) * HIDSZ + n0 + 32 + l];
    b.y = enc_w[(kk + 1) * HIDSZ + n0 + 32 + l];
    c2 = wmma_f32(a, b, c2);
    b.x = enc_w[(kk + 0) * HIDSZ + n0 + 48 + l];
    b.y = enc_w[(kk + 1) * HIDSZ + n0 + 48 + l];
    c3 = wmma_f32(a, b, c3);
  }

  v8f acc[4] = {c0, c1, c2, c3};
#pragma unroll
  for (int j = 0; j < 4; ++j) {
#pragma unroll
    for (int i = 0; i < 8; ++i) {
      const int row = m0 + i + 8 * up;
      const int col = n0 + 16 * j + l;
      float v = acc[j][i] + enc_b[col];
      const float* ci = comm_in + (size_t)row * COMMSZ + col;
#pragma unroll
      for (int a = 0; a < NAGENTS; ++a) v += ci[a * HIDSZ];
      pre[row * HIDSZ + col] = v;
    }
  }
}

// --------------------------------------------------------------------------
// Kernel 3: hid = relu(pre + prev_hid @ rnn_w[id] + rnn_b[id])
// Grouped GEMM: block = (model m, 64-col group); padded 16-row M tiles.
// --------------------------------------------------------------------------
__global__ __launch_bounds__(32) void rnn_kernel(
    const float* __restrict__ prev_hid,  // [BN, HIDSZ]
    const float* __restrict__ rnn_w,     // [NMODELS, HIDSZ, HIDSZ]
    const float* __restrict__ rnn_b,     // [NMODELS, HIDSZ]
    const float* __restrict__ pre,       // [BN, HIDSZ]
    const int* __restrict__ counts,
    const int* __restrict__ rowidx,
    float* __restrict__ hid)             // [BN, HIDSZ] (inside d_out)
{
  const int m  = blockIdx.x;             // model
  const int n0 = blockIdx.y << 6;        // output col group (64 wide)
  const int cnt = counts[m];
  const int* rows = rowidx + m * BN;
  const int lane = threadIdx.x;
  const int up = lane >> 4;
  const int l  = lane & 15;
  const float* W = rnn_w + (size_t)m * HIDSZ * HIDSZ;

  for (int mt = 0; mt < cnt; mt += 16) {
    const int ri  = mt + l;              // this lane's A row within group
    const int row = (ri < cnt) ? rows[ri] : -1;
    v8f c0 = {}, c1 = {}, c2 = {}, c3 = {};
#pragma unroll 4
    for (int k = 0; k < HIDSZ; k += 4) {
      const int kk = k + 2 * up;
      if ((k & 31) == 0 && k + 32 < HIDSZ)
        prefetch_wgp(&W[(kk + 32) * HIDSZ + n0]);
      v2f a;
      a.x = (row >= 0) ? prev_hid[row * HIDSZ + kk + 0] : 0.f;
      a.y = (row >= 0) ? prev_hid[row * HIDSZ + kk + 1] : 0.f;
      v2f b;
      b.x = W[(kk + 0) * HIDSZ + n0 + l];
      b.y = W[(kk + 1) * HIDSZ + n0 + l];
      c0 = wmma_f32(a, b, c0);
      b.x = W[(kk + 0) * HIDSZ + n0 + 16 + l];
      b.y = W[(kk + 1) * HIDSZ + n0 + 16 + l];
      c1 = wmma_f32(a, b, c1);
      b.x = W[(kk + 0) * HIDSZ + n0 + 32 + l];
      b.y = W[(kk + 1) * HIDSZ + n0 + 32 + l];
      c2 = wmma_f32(a, b, c2);
      b.x = W[(kk + 0) * HIDSZ + n0 + 48 + l];
      b.y = W[(kk + 1) * HIDSZ + n0 + 48 + l];
      c3 = wmma_f32(a, b, c3);
    }
    v8f acc[4] = {c0, c1, c2, c3};
#pragma unroll
    for (int j = 0; j < 4; ++j) {
#pragma unroll
      for (int i = 0; i < 8; ++i) {
        const int gi = mt + i + 8 * up;
        if (gi < cnt) {
          const int r = rows[gi];
          const int col = n0 + 16 * j + l;
          float v = acc[j][i] + pre[r * HIDSZ + col] + rnn_b[m * HIDSZ + col];
          hid[r * HIDSZ + col] = fmaxf(v, 0.f);
        }
      }
    }
  }
}

// --------------------------------------------------------------------------
// Kernel 4: heads. blockIdx.y == 0 -> act logits (one 16-col subtile);
//                  1..32 -> comm_out 64-col groups.
// --------------------------------------------------------------------------
__global__ __launch_bounds__(32) void heads_kernel(
    const float* __restrict__ hid,      // [BN, HIDSZ]
    const float* __restrict__ act_w,    // [NMODELS, HIDSZ, NACTIONS]
    const float* __restrict__ act_b,    // [NMODELS, NACTIONS]
    const float* __restrict__ comm_w,   // [NMODELS, HIDSZ, COMMSZ]
    const float* __restrict__ comm_b,   // [NMODELS, COMMSZ]
    const int* __restrict__ counts,
    const int* __restrict__ rowidx,
    float* __restrict__ logits,         // [BN, NACTIONS] workspace
    float* __restrict__ comm_out)       // [BN, COMMSZ] (inside d_out)
{
  const int m  = blockIdx.x;
  const int nt = blockIdx.y;             // 0 = act, 1..32 = comm groups
  const int cnt = counts[m];
  const int* rows = rowidx + m * BN;
  const int lane = threadIdx.x;
  const int up = lane >> 4;
  const int l  = lane & 15;
  const bool is_act = (nt == 0);
  const int nsub = is_act ? 1 : 4;
  const int n0 = is_act ? 0 : ((nt - 1) << 6);
  const int NOUT = is_act ? NACTIONS : COMMSZ;
  const float* W = is_act ? (act_w + (size_t)m * HIDSZ * NACTIONS)
                          : (comm_w + (size_t)m * HIDSZ * COMMSZ);
  const float* B = is_act ? (act_b + (size_t)m * NACTIONS)
                          : (comm_b + (size_t)m * COMMSZ);

  for (int mt = 0; mt < cnt; mt += 16) {
    const int ri  = mt + l;
    const int row = (ri < cnt) ? rows[ri] : -1;
    v8f acc[4] = {{}, {}, {}, {}};
#pragma unroll 4
    for (int k = 0; k < HIDSZ; k += 4) {
      const int kk = k + 2 * up;
      if ((k & 31) == 0 && k + 32 < HIDSZ)
        prefetch_wgp(&W[(size_t)(kk + 32) * NOUT + n0]);
      v2f a;
      a.x = (row >= 0) ? hid[row * HIDSZ + kk + 0] : 0.f;
      a.y = (row >= 0) ? hid[row * HIDSZ + kk + 1] : 0.f;
#pragma unroll
      for (int j = 0; j < 4; ++j) {
        if (j < nsub) {
          v2f b;
          b.x = W[(size_t)(kk + 0) * NOUT + n0 + 16 * j + l];
          b.y = W[(size_t)(kk + 1) * NOUT + n0 + 16 * j + l];
          acc[j] = wmma_f32(a, b, acc[j]);
        }
      }
    }
#pragma unroll
    for (int j = 0; j < 4; ++j) {
      if (j < nsub) {
#pragma unroll
        for (int i = 0; i < 8; ++i) {
          const int gi = mt + i + 8 * up;
          if (gi < cnt) {
            const int r = rows[gi];
            const int col = n0 + 16 * j + l;
            const float v = acc[j][i] + B[col];
            if (is_act) logits[r * NACTIONS + col] = v;
            else        comm_out[(size_t)r * COMMSZ + col] = v;
          }
        }
      }
    }
  }
}

// --------------------------------------------------------------------------
// Kernel 5: baseline matvec + 16-wide softmax. One wave per row.
// --------------------------------------------------------------------------
__global__ __launch_bounds__(32) void tail_kernel(
    const float* __restrict__ hid,      // [BN, HIDSZ]
    const float* __restrict__ base_w,   // [NMODELS, HIDSZ]
    const float* __restrict__ base_b,   // [NMODELS]
    const int* __restrict__ ids,
    const float* __restrict__ logits,   // [BN, NACTIONS]
    float* __restrict__ action_prob,    // [BN, NACTIONS]
    float* __restrict__ baseline)       // [BN]
{
  const int row = blockIdx.x;
  const int lane = threadIdx.x;
  const int id = ids[row];

  // baseline = hid[row] . base_w[id] + base_b[id]
  float s = 0.f;
  for (int k = lane; k < HIDSZ; k += 32)
    s += hid[row * HIDSZ + k] * base_w[(size_t)id * HIDSZ + k];
#pragma unroll
  for (int off = 16; off > 0; off >>= 1) s += __shfl_down(s, off, 32);
  if (lane == 0) baseline[row] = s + base_b[id];

  // softmax over 16 logits (lanes 0..15 active; xor offsets <=8 stay in half)
  float x = (lane < NACTIONS) ? logits[row * NACTIONS + lane] : -3.0e38f;
  float mx = x;
#pragma unroll
  for (int off = 8; off > 0; off >>= 1) mx = fmaxf(mx, __shfl_xor(mx, off, 32));
  float e = (lane < NACTIONS) ? __expf(x - mx) : 0.f;
  float sum = e;
#pragma unroll
  for (int off = 8; off > 0; off >>= 1) sum += __shfl_xor(sum, off, 32);
  if (lane < NACTIONS) action_prob[row * NACTIONS + lane] = e / sum;
}

// --------------------------------------------------------------------------
extern "C" void kernel_launch(void* const* d_in, const int* in_sizes, int n_in,
                              void* d_out, int out_size, void* d_ws, size_t ws_size,
                              hipStream_t stream) {
  const float* inp      = (const float*)d_in[0];   // [256,512]
  const float* prev_hid = (const float*)d_in[1];   // [256,256]
  const float* comm_in  = (const float*)d_in[2];   // [256,8,256]
  const int*   ids      = (const int*)  d_in[3];   // [256]
  const float* enc_w    = (const float*)d_in[4];
  const float* enc_b    = (const float*)d_in[5];
  const float* rnn_w    = (const float*)d_in[6];
  const float* rnn_b    = (const float*)d_in[7];
  const float* act_w    = (const float*)d_in[8];
  const float* act_b    = (const float*)d_in[9];
  const float* base_w   = (const float*)d_in[10];
  const float* base_b   = (const float*)d_in[11];
  const float* comm_w   = (const float*)d_in[12];
  const float* comm_b   = (const float*)d_in[13];

  // d_out layout (return order, flat): action_prob, baseline, hid, comm_out
  float* out_action = (float*)d_out;                       // 256*16
  float* out_base   = out_action + BN * NACTIONS;          // 256
  float* out_hid    = out_base + BN;                       // 256*256
  float* out_comm   = out_hid + BN * HIDSZ;                // 256*2048

  // workspace: counts | rowidx | pre | logits
  char* ws = (char*)d_ws;
  int*   counts = (int*)ws;                                 // 64
  int*   rowidx = (int*)(ws + 256);                         // 64*256
  float* pre    = (float*)(ws + 256 + 65536);               // 256*256
  float* logits = (float*)(ws + 256 + 65536 + 262144);      // 256*16

  group_kernel<<<1, BN, 0, stream>>>(ids, counts, rowidx);
  enc_pre_kernel<<<64, 32, 0, stream>>>(inp, enc_w, enc_b, comm_in, pre);
  rnn_kernel<<<dim3(NMODELS, HIDSZ / 64), 32, 0, stream>>>(
      prev_hid, rnn_w, rnn_b, pre, counts, rowidx, out_hid);
  heads_kernel<<<dim3(NMODELS, 1 + COMMSZ / 64), 32, 0, stream>>>(
      out_hid, act_w, act_b, comm_w, comm_b, counts, rowidx, logits, out_comm);
  tail_kernel<<<BN, 32, 0, stream>>>(out_hid, base_w, base_b, ids, logits,
                                     out_action, out_base);
}